// Decoder_40157944217903
// MI455X (gfx1250) — compile-verified
//
#include <hip/hip_runtime.h>
#include <stdint.h>

#define B_ 4
#define N_ 1500
#define L_ 81
#define K_ 200
#define BLK_ (L_ * K_)          // 16200 candidates per batch
#define SCORE_T 0.7f
#define IOU_T 0.5f

typedef unsigned int v4u __attribute__((ext_vector_type(4)));
typedef int          v8i __attribute__((ext_vector_type(8)));
typedef int          v4i __attribute__((ext_vector_type(4)));

// ---------------------------------------------------------------------------
// Tensor Data Mover: 1-D contiguous tile (nelem4 x 4-byte elements) global->LDS.
// D# built per CDNA5 ISA:
//   group0: count=1 | lds_addr | global_addr[56:0] | type=2
//   group1: data_size=2 (4B), tensor_dim0=tile_dim0=nelem4, tensor_dim1=1
// ---------------------------------------------------------------------------
__device__ __forceinline__ void tdm_load_1d(uint32_t lds_addr, const void* gptr,
                                            uint32_t nelem4) {
  uint64_t ga = (uint64_t)(uintptr_t)gptr;
  v4u g0;
  g0.x = 1u;                                            // count=1, user mode
  g0.y = lds_addr;                                      // LDS byte address
  g0.z = (uint32_t)(ga & 0xFFFFFFFFu);                  // global_addr[31:0]
  g0.w = (uint32_t)((ga >> 32) & 0x01FFFFFFu) | (2u << 30); // [56:32] | type=2
  v8i g1;
  g1[0] = (int)(2u << 16);                  // wg_mask=0, data_size=2 (4 bytes)
  g1[1] = (int)((nelem4 & 0xFFFFu) << 16);  // barrier_addr=0 | tensor_dim0[15:0]
  g1[2] = (int)((nelem4 >> 16) | (1u << 16)); // tensor_dim0[31:16] | tensor_dim1=1
  g1[3] = (int)((nelem4 & 0xFFFFu) << 16);  // tensor_dim1[31:16]=0 | tile_dim0
  g1[4] = 0;                                // tile_dim1=0, tile_dim2=0 (1-D)
  g1[5] = (int)nelem4;                      // tensor_dim0_stride[31:0]
  g1[6] = 0;                                // stride hi | tensor_dim1_stride lo
  g1[7] = 0;
  v4i z = {0, 0, 0, 0};
#if __clang_major__ >= 23
  v8i z8 = {0, 0, 0, 0, 0, 0, 0, 0};
  __builtin_amdgcn_tensor_load_to_lds(g0, g1, z, z, z8, 0);
#else
  __builtin_amdgcn_tensor_load_to_lds(g0, g1, z, z, 0);
#endif
}

// ---------------------------------------------------------------------------
// Kernel 0: zero-init outputs + candidate counters (graph-safe, deterministic)
// ---------------------------------------------------------------------------
__global__ void k_init(float* __restrict__ out, int* __restrict__ candCount) {
  int t = blockIdx.x * blockDim.x + threadIdx.x;
  if (t < B_ * K_ * 6) out[t] = 0.0f;   // 4800 floats: boxes+labels+scores
  if (t < B_) candCount[t] = 0;
}

// ---------------------------------------------------------------------------
// Kernel 1: per-ROI argmax over 81 classes -> background flag
// (jnp.argmax keeps the first max on ties -> strict '>' keeps lowest index)
// ---------------------------------------------------------------------------
__global__ void k_argmax_bg(const float* __restrict__ probs, int* __restrict__ bg) {
  int t = blockIdx.x * blockDim.x + threadIdx.x;
  if (t >= B_ * N_) return;
  const float* p = probs + (size_t)t * L_;
  float best = p[0];
  int bi = 0;
  for (int l = 1; l < L_; ++l) {
    float v = p[l];
    if (v > best) { best = v; bi = l; }
  }
  bg[t] = (bi == 0) ? 1 : 0;
}

// ---------------------------------------------------------------------------
// Kernel 2: delta->bbox decode for all (b,l,n); class-major [B,L,N,4] layout
// so each NMS workgroup reads one contiguous 24 KB tile.
// ---------------------------------------------------------------------------
__global__ void k_decode(const float* __restrict__ rois,
                         const float* __restrict__ deltas,
                         const float* __restrict__ probs,
                         const int* __restrict__ bg,
                         float4* __restrict__ boxes,
                         float* __restrict__ scores) {
  int t = blockIdx.x * blockDim.x + threadIdx.x;   // t = (b*L + l)*N + n
  if (t >= B_ * L_ * N_) return;
  int n = t % N_;
  int bl = t / N_;
  int l = bl % L_;
  int b = bl / L_;
  int bn = b * N_ + n;

  const float* r = rois + (size_t)bn * 4;
  float y1 = r[0], x1 = r[1], y2 = r[2], x2 = r[3];
  float h = y2 - y1, w = x2 - x1;
  float cy = y1 + 0.5f * h, cx = x1 + 0.5f * w;

  const float* d = deltas + ((size_t)bn * L_ + l) * 4;
  float dy = d[0] * 0.1f, dx = d[1] * 0.1f;
  float dh = d[2] * 0.2f, dw = d[3] * 0.2f;

  float bh = expf(dh) * h, bw = expf(dw) * w;
  float bcy = dy * h + cy, bcx = dx * w + cx;
  float oy1 = bcy - 0.5f * bh, ox1 = bcx - 0.5f * bw;

  boxes[t] = make_float4(oy1, ox1, oy1 + bh, ox1 + bw);
  scores[t] = bg[bn] ? 0.0f : probs[(size_t)bn * L_ + l];
}

// ---------------------------------------------------------------------------
// Kernel 3: greedy NMS, one workgroup per (b,l). Boxes+scores TDM-DMA'd into
// LDS; per-iteration argmax via wave32 shuffle + cross-wave LDS reduce.
// ---------------------------------------------------------------------------
__global__ void __launch_bounds__(256) k_nms(const float* __restrict__ boxes,
                                             const float* __restrict__ scores,
                                             int* __restrict__ selIdx,
                                             float* __restrict__ selScore) {
  __shared__ alignas(16) float4 s_box[N_];   // 24000 B
  __shared__ alignas(16) float s_sc[N_];     //  6000 B
  __shared__ float s_rv[8];
  __shared__ int s_ri[8];
  __shared__ float s_bestV;
  __shared__ int s_bestI;

  const int tid = threadIdx.x;
  const int bl = blockIdx.x;                 // b*L + l
  const float4* gbox = (const float4*)boxes + (size_t)bl * N_;
  const float* gsc = scores + (size_t)bl * N_;
  int* oIdx = selIdx + (size_t)bl * K_;
  float* oSc = selScore + (size_t)bl * K_;

  // Prefill: invalid slots match reference's (idx=-1, score=0) tail.
  for (int k = tid; k < K_; k += 256) { oIdx[k] = -1; oSc[k] = 0.0f; }

  // Async DMA stage via Tensor Data Mover (wave 0 issues; TENSORcnt is
  // per-wave, so wave 0 waits, then the workgroup barrier publishes LDS).
  if (tid < 32) {
    tdm_load_1d((uint32_t)(uintptr_t)&s_box[0], (const void*)gbox, N_ * 4u);
    tdm_load_1d((uint32_t)(uintptr_t)&s_sc[0], (const void*)gsc, (uint32_t)N_);
    __builtin_amdgcn_s_wait_tensorcnt(0);
  }
  __syncthreads();

  const float NEG = -__builtin_huge_valf();
  for (int k = 0; k < K_; ++k) {
    // --- argmax over LDS scores (first-max tie-break = lowest index) ---
    float bv = NEG;
    int bi = 0x7FFFFFFF;
    for (int i = tid; i < N_; i += 256) {
      float v = s_sc[i];
      if (v > bv) { bv = v; bi = i; }      // per-thread indices increase
    }
    for (int off = 16; off > 0; off >>= 1) {
      float ov = __shfl_down(bv, off, 32);
      int oi = __shfl_down(bi, off, 32);
      if (ov > bv || (ov == bv && oi < bi)) { bv = ov; bi = oi; }
    }
    if ((tid & 31) == 0) { s_rv[tid >> 5] = bv; s_ri[tid >> 5] = bi; }
    __syncthreads();
    if (tid == 0) {
      float fv = s_rv[0];
      int fi = s_ri[0];
      for (int wv = 1; wv < 8; ++wv) {
        float ov = s_rv[wv]; int oi = s_ri[wv];
        if (ov > fv || (ov == fv && oi < fi)) { fv = ov; fi = oi; }
      }
      s_bestV = fv; s_bestI = fi;
    }
    __syncthreads();

    float s = s_bestV;
    int j = s_bestI;
    if (!(s > SCORE_T)) break;             // remaining slots stay (-1, 0)
    if (tid == 0) { oIdx[k] = j; oSc[k] = s; }

    // --- suppress IoU > 0.5 (includes j itself: IoU=1) ---
    float4 bb = s_box[j];
    float ba = (bb.z - bb.x) * (bb.w - bb.y);
    for (int i = tid; i < N_; i += 256) {
      float4 o = s_box[i];
      float yy1 = fmaxf(bb.x, o.x), xx1 = fmaxf(bb.y, o.y);
      float yy2 = fminf(bb.z, o.z), xx2 = fminf(bb.w, o.w);
      float inter = fmaxf(yy2 - yy1, 0.0f) * fmaxf(xx2 - xx1, 0.0f);
      float oa = (o.z - o.x) * (o.w - o.y);
      float uni = ba + oa - inter;
      float iou = (uni > 0.0f) ? (inter / uni) : 0.0f;
      if (iou > IOU_T) s_sc[i] = NEG;
    }
    __syncthreads();
  }
}

// ---------------------------------------------------------------------------
// Kernel 4: compact positive candidates per batch (score>0 <=> valid > 0.7)
// ---------------------------------------------------------------------------
__global__ void k_compact(const float* __restrict__ selScore,
                          int* __restrict__ candCount,
                          float* __restrict__ candScore,
                          int* __restrict__ candFlat) {
  int t = blockIdx.x * blockDim.x + threadIdx.x;  // over B*L*K
  if (t >= B_ * BLK_) return;
  float s = selScore[t];
  if (s > 0.0f) {
    int b = t / BLK_;
    int f = t - b * BLK_;                         // flat index l*K + k
    int pos = atomicAdd(&candCount[b], 1);
    candScore[(size_t)b * BLK_ + pos] = s;
    candFlat[(size_t)b * BLK_ + pos] = f;
  }
}

// ---------------------------------------------------------------------------
// Kernel 5: per-batch top-200 via stable rank (score desc, flat idx asc ==
// lax.top_k tie order); gather clipped boxes / labels / scores.
// ---------------------------------------------------------------------------
__global__ void __launch_bounds__(256) k_topk(const float* __restrict__ candScore,
                                              const int* __restrict__ candFlat,
                                              const int* __restrict__ candCount,
                                              const int* __restrict__ selIdx,
                                              const float4* __restrict__ boxes,
                                              float* __restrict__ out) {
  int b = blockIdx.x;
  int tid = threadIdx.x;
  float4* oBox = (float4*)out;                   // [B,200,4]
  float* oLab = out + B_ * K_ * 4;               // [B,200]
  float* oSc = out + B_ * K_ * 4 + B_ * K_;      // [B,200]

  int cnt = candCount[b];
  if (cnt > BLK_) cnt = BLK_;
  const float* cs = candScore + (size_t)b * BLK_;
  const int* cf = candFlat + (size_t)b * BLK_;

  for (int i = tid; i < cnt; i += 256) {
    float si = cs[i];
    int fi = cf[i];
    int rank = 0;
    for (int jx = 0; jx < cnt; ++jx) {
      float sj = cs[jx];
      int fj = cf[jx];
      rank += (sj > si) || (sj == si && fj < fi);
    }
    if (rank < K_) {
      int l = fi / K_;
      int k = fi - l * K_;
      int j = selIdx[((size_t)b * L_ + l) * K_ + k];
      float4 bb = boxes[((size_t)b * L_ + l) * N_ + j];
      float4 cb;
      cb.x = fminf(fmaxf(bb.x, 0.0f), 1.0f);
      cb.y = fminf(fmaxf(bb.y, 0.0f), 1.0f);
      cb.z = fminf(fmaxf(bb.z, 0.0f), 1.0f);
      cb.w = fminf(fmaxf(bb.w, 0.0f), 1.0f);
      oBox[b * K_ + rank] = cb;
      oLab[b * K_ + rank] = (float)l;
      oSc[b * K_ + rank] = si;
    }
  }
}

// ---------------------------------------------------------------------------
extern "C" void kernel_launch(void* const* d_in, const int* in_sizes, int n_in,
                              void* d_out, int out_size, void* d_ws, size_t ws_size,
                              hipStream_t stream) {
  const float* rois = (const float*)d_in[0];    // [B,N,4]
  const float* deltas = (const float*)d_in[1];  // [B,N,L*4]
  const float* probs = (const float*)d_in[2];   // [B,N,L]
  float* out = (float*)d_out;

  char* ws = (char*)d_ws;
  size_t off = 0;
  auto alloc = [&](size_t bytes) -> void* {
    off = (off + 255) & ~(size_t)255;
    void* p = ws + off;
    off += bytes;
    return p;
  };
  float4* boxes   = (float4*)alloc((size_t)B_ * L_ * N_ * 4 * sizeof(float));
  float* scores   = (float*)alloc((size_t)B_ * L_ * N_ * sizeof(float));
  int* bg         = (int*)alloc((size_t)B_ * N_ * sizeof(int));
  int* selIdx     = (int*)alloc((size_t)B_ * BLK_ * sizeof(int));
  float* selScore = (float*)alloc((size_t)B_ * BLK_ * sizeof(float));
  float* candScore= (float*)alloc((size_t)B_ * BLK_ * sizeof(float));
  int* candFlat   = (int*)alloc((size_t)B_ * BLK_ * sizeof(int));
  int* candCount  = (int*)alloc((size_t)B_ * sizeof(int));
  (void)ws_size; (void)in_sizes; (void)n_in; (void)out_size;

  k_init<<<(B_ * K_ * 6 + 255) / 256, 256, 0, stream>>>(out, candCount);
  k_argmax_bg<<<(B_ * N_ + 255) / 256, 256, 0, stream>>>(probs, bg);
  k_decode<<<(B_ * L_ * N_ + 255) / 256, 256, 0, stream>>>(
      rois, deltas, probs, bg, boxes, scores);
  k_nms<<<B_ * L_, 256, 0, stream>>>((const float*)boxes, scores, selIdx, selScore);
  k_compact<<<(B_ * BLK_ + 255) / 256, 256, 0, stream>>>(
      selScore, candCount, candScore, candFlat);
  k_topk<<<B_, 256, 0, stream>>>(candScore, candFlat, candCount, selIdx,
                                 (const float4*)boxes, out);
}